// FusedChaosSSMBlock_45329084842041
// MI455X (gfx1250) — compile-verified
//
#include <hip/hip_runtime.h>
#include <hip/hip_bf16.h>

typedef __attribute__((ext_vector_type(2))) float v2f;
typedef __attribute__((ext_vector_type(8))) float v8f;

#define CHUNKS 64
#define CLEN   64   // T = CHUNKS * CLEN = 4096

// ---------------------------------------------------------------------------
// RMSNorm: one block per row of length D=1024, 256 threads * float4.
// ---------------------------------------------------------------------------
__global__ __launch_bounds__(256) void rmsnorm_kernel(
    const float* __restrict__ x, const float* __restrict__ w,
    float* __restrict__ out, int D)
{
  __shared__ float red[8];
  const int row = blockIdx.x;
  const int tid = threadIdx.x;
  const float* xr = x + (size_t)row * D;

  float4 v = *(const float4*)(xr + tid * 4);
  float s = v.x * v.x + v.y * v.y + v.z * v.z + v.w * v.w;
  // wave32 reduce
  #pragma unroll
  for (int off = 16; off > 0; off >>= 1) s += __shfl_xor(s, off, 32);
  if ((tid & 31) == 0) red[tid >> 5] = s;
  __syncthreads();
  float tot = red[0] + red[1] + red[2] + red[3] + red[4] + red[5] + red[6] + red[7];
  float scale = rsqrtf(tot / (float)D + 1e-6f);

  float4 wv = *(const float4*)(w + tid * 4);
  float4 o;
  o.x = v.x * scale * wv.x;
  o.y = v.y * scale * wv.y;
  o.z = v.z * scale * wv.z;
  o.w = v.w * scale * wv.w;
  *(float4*)(out + (size_t)row * D + tid * 4) = o;
}

// ---------------------------------------------------------------------------
// Chunked scan, pass 1: local scan inside each chunk of CLEN timesteps.
// ---------------------------------------------------------------------------
__global__ __launch_bounds__(256) void scan_local_kernel(
    const float* __restrict__ normed, const float* __restrict__ a_log,
    float* __restrict__ hloc, float* __restrict__ S, int T, int D)
{
  const int d = blockIdx.x * 256 + threadIdx.x;
  const int c = blockIdx.y;
  const int b = blockIdx.z;
  const float a = __expf(-__expf(a_log[d]));
  const size_t base = ((size_t)b * T + (size_t)c * CLEN) * D + d;
  float h = 0.0f;
  for (int t = 0; t < CLEN; ++t) {
    size_t idx = base + (size_t)t * D;
    h = a * h + normed[idx];
    hloc[idx] = h;
  }
  S[((size_t)b * CHUNKS + c) * D + d] = h;
}

// ---------------------------------------------------------------------------
// Chunked scan, pass 2: per-channel serial combine over 64 chunk summaries.
// ---------------------------------------------------------------------------
__global__ __launch_bounds__(256) void scan_carry_kernel(
    const float* __restrict__ S, const float* __restrict__ a_log,
    float* __restrict__ carry, int D)
{
  const int gid = blockIdx.x * 256 + threadIdx.x; // over B*D
  const int b = gid / D;
  const int d = gid - b * D;
  float a = __expf(-__expf(a_log[d]));
  float P = a;
  #pragma unroll
  for (int i = 0; i < 6; ++i) P *= P;  // a^64
  float cur = 0.0f;
  for (int c = 0; c < CHUNKS; ++c) {
    size_t idx = ((size_t)b * CHUNKS + c) * D + d;
    carry[idx] = cur;
    cur = P * cur + S[idx];
  }
}

// ---------------------------------------------------------------------------
// Chunked scan, pass 3: h_t = a^(t+1)*carry + h_local, then fused
// x1 = x + h*c_out + normed*d_skip.
// ---------------------------------------------------------------------------
__global__ __launch_bounds__(256) void scan_apply_kernel(
    const float* __restrict__ x, const float* __restrict__ normed,
    const float* __restrict__ hloc, const float* __restrict__ carry,
    const float* __restrict__ a_log, const float* __restrict__ c_out,
    const float* __restrict__ d_skip, float* __restrict__ x1, int T, int D)
{
  const int d = blockIdx.x * 256 + threadIdx.x;
  const int c = blockIdx.y;
  const int b = blockIdx.z;
  const float a  = __expf(-__expf(a_log[d]));
  const float cw = c_out[d];
  const float dw = d_skip[d];
  const float cin = carry[((size_t)b * CHUNKS + c) * D + d];
  const size_t base = ((size_t)b * T + (size_t)c * CLEN) * D + d;
  float pw = a;
  for (int t = 0; t < CLEN; ++t) {
    size_t idx = base + (size_t)t * D;
    float h = pw * cin + hloc[idx];
    pw *= a;
    x1[idx] = x[idx] + h * cw + normed[idx] * dw;
  }
}

// ---------------------------------------------------------------------------
// fp32 WMMA GEMM:  C[M,N] = act( A[M,K] * W[N,K]^T ) (+ residual)
// Block: 256 threads = 8 waves. Block tile 128(M) x 128(N). Waves 4x2, each
// wave owns 32x64 = 2x4 accumulators of V_WMMA_F32_16X16X4_F32.
//
// Staging: ping-pong LDS double buffer filled by GLOBAL_LOAD_ASYNC_TO_LDS_B128
// (no VGPR round trip; tracked with ASYNCcnt). Each thread issues 4 async
// B128 loads per stage; s_wait_asynccnt 0x4 retires only the current stage
// (async loads complete in order per wave), so the next stage's loads stay in
// flight behind the 32 WMMAs. A and W staged row-major, stride 20 floats
// (conflict-free; rows are 80B so B128 writes stay 16B-aligned). Fragments
// are single ds_load_b64 each per ISA 7.12.2 layouts.
// ---------------------------------------------------------------------------
#define LDA 20

__global__ __launch_bounds__(256) void gemm_wmma_f32(
    const float* __restrict__ A, const float* __restrict__ W,
    const float* __restrict__ res, float* __restrict__ C,
    int M, int K, int N, int act_silu)
{
  __shared__ float As[2][128 * LDA];
  __shared__ float Ws[2][128 * LDA];

  const int tid  = threadIdx.x;
  const int lane = tid & 31;
  const int wid  = tid >> 5;
  const int wm   = (wid & 3) * 32;   // wave M offset in block tile (4 waves)
  const int wn   = (wid >> 2) * 64;  // wave N offset in block tile (2 waves)
  const int blockN = blockIdx.x * 128;
  const int blockM = blockIdx.y * 128;

  const int lh = lane & 15;          // 0..15
  const int lq = lane >> 4;          // lane half: 0 or 1

  const int qa = tid & 3;            // quad within a 16-float row
  const int ra = tid >> 2;           // 0..63

  const float* Ap = A + (size_t)(blockM + ra) * K + qa * 4;
  const float* Wp = W + (size_t)(blockN + ra) * K + qa * 4;
  const size_t rowK64 = (size_t)64 * K;

  // LDS byte addresses for this thread's 4 async destinations (buffer 0).
  // Generic->LDS: low 32 bits of the generic pointer are the LDS address.
  const unsigned laA0 = (unsigned)(size_t)&As[0][ra * LDA + qa * 4];
  const unsigned laA1 = (unsigned)(size_t)&As[0][(ra + 64) * LDA + qa * 4];
  const unsigned laW0 = (unsigned)(size_t)&Ws[0][ra * LDA + qa * 4];
  const unsigned laW1 = (unsigned)(size_t)&Ws[0][(ra + 64) * LDA + qa * 4];
  const unsigned bufB = (unsigned)(128 * LDA * sizeof(float)); // bytes per buffer

  #define ASYNC_B128(lds, gptr)                                            \
    asm volatile("global_load_async_to_lds_b128 %0, %1, off"               \
                 :: "v"(lds), "v"((unsigned long long)(size_t)(gptr))      \
                 : "memory")

  #define STAGE(buf, koff) do {                                            \
    unsigned o = (buf) * bufB;                                             \
    ASYNC_B128(laA0 + o, Ap + (koff));                                     \
    ASYNC_B128(laA1 + o, Ap + rowK64 + (koff));                            \
    ASYNC_B128(laW0 + o, Wp + (koff));                                     \
    ASYNC_B128(laW1 + o, Wp + rowK64 + (koff));                            \
  } while (0)

  v8f acc[2][4] = {};

  STAGE(0, 0);  // prologue: stage 0 in flight

  const int nIter = K / 16;
  for (int i = 0; i < nIter; ++i) {
    const int cur = i & 1;
    const bool hasNext = (i + 1 < nIter);
    if (hasNext) {
      STAGE(cur ^ 1, (i + 1) * 16);
      // retire current stage's 4 loads; keep next stage's 4 in flight
      asm volatile("s_wait_asynccnt 0x4" ::: "memory");
    } else {
      asm volatile("s_wait_asynccnt 0x0" ::: "memory");
    }
    __syncthreads();   // all waves' loads for `cur` are now in LDS

    const float* Asb = &As[cur][0];
    const float* Wsb = &Ws[cur][0];

    // load all fragments for this 16-wide K block
    v2f afrag[4][2];
    v2f bfrag[4][4];
    #pragma unroll
    for (int s = 0; s < 4; ++s) {
      const int ka = s * 4 + lq * 2;   // {ka, ka+1} per ISA 7.12.2 layout
      #pragma unroll
      for (int mi = 0; mi < 2; ++mi)
        afrag[s][mi] = *(const v2f*)(&Asb[(wm + mi * 16 + lh) * LDA + ka]);
      #pragma unroll
      for (int nj = 0; nj < 4; ++nj)
        bfrag[s][nj] = *(const v2f*)(&Wsb[(wn + nj * 16 + lh) * LDA + ka]);
    }

    // 32 back-to-back WMMAs; acc reuse distance = 8
    #pragma unroll
    for (int s = 0; s < 4; ++s)
      #pragma unroll
      for (int mi = 0; mi < 2; ++mi)
        #pragma unroll
        for (int nj = 0; nj < 4; ++nj)
          acc[mi][nj] = __builtin_amdgcn_wmma_f32_16x16x4_f32(
              false, afrag[s][mi], false, bfrag[s][nj],
              (short)0, acc[mi][nj], false, false);

    __syncthreads();   // protect buffer `cur` before stage i+2 overwrites it
  }

  #undef STAGE
  #undef ASYNC_B128

  // epilogue: C/D layout -> VGPR r holds M = r + 8*lane_half, N = lane%16
  #pragma unroll
  for (int mi = 0; mi < 2; ++mi) {
    #pragma unroll
    for (int nj = 0; nj < 4; ++nj) {
      const int col = blockN + wn + nj * 16 + lh;
      #pragma unroll
      for (int r = 0; r < 8; ++r) {
        const int row = blockM + wm + mi * 16 + lq * 8 + r;
        float v = acc[mi][nj][r];
        if (act_silu) v = v / (1.0f + __expf(-v));   // SiLU
        const size_t idx = (size_t)row * N + col;
        if (res) v += res[idx];
        C[idx] = v;
      }
    }
  }
}

// ---------------------------------------------------------------------------
// Launch
// ---------------------------------------------------------------------------
extern "C" void kernel_launch(void* const* d_in, const int* in_sizes, int n_in,
                              void* d_out, int out_size, void* d_ws, size_t ws_size,
                              hipStream_t stream) {
  (void)in_sizes; (void)n_in; (void)out_size; (void)ws_size;

  const float* x       = (const float*)d_in[0];
  const float* in_w    = (const float*)d_in[1];
  const float* a_log   = (const float*)d_in[2];
  const float* c_out   = (const float*)d_in[3];
  const float* d_skip  = (const float*)d_in[4];
  const float* ffn_w   = (const float*)d_in[5];
  const float* fc_w    = (const float*)d_in[6];  // (H, D)
  const float* proj_w  = (const float*)d_in[7];  // (D, H)
  float* out = (float*)d_out;

  const int B = 4, T = 4096, D = 1024, H = 2048;
  const int M = B * T;  // 16384 rows

  // Workspace layout (bytes):
  //  [0, M*H*4)                : hidden (GEMM1 out) -- first M*D*4 doubles as `normed`
  //  [M*H*4, +M*D*4)           : hloc, later reused as n2
  //  [.., +M*D*4)              : x1
  //  [.., +B*CHUNKS*D*4)       : chunk summaries S
  //  [.., +B*CHUNKS*D*4)       : carries
  char* ws = (char*)d_ws;
  float* hidden = (float*)ws;
  float* normed = hidden;                      // dead before GEMM1 writes hidden
  float* buf1   = (float*)(ws + (size_t)M * H * 4);            // hloc -> n2
  float* x1     = (float*)(ws + (size_t)M * H * 4 + (size_t)M * D * 4);
  float* S      = (float*)(ws + (size_t)M * H * 4 + 2ull * M * D * 4);
  float* carry  = S + (size_t)B * CHUNKS * D;

  rmsnorm_kernel<<<M, 256, 0, stream>>>(x, in_w, normed, D);

  scan_local_kernel<<<dim3(D / 256, CHUNKS, B), 256, 0, stream>>>(
      normed, a_log, buf1, S, T, D);
  scan_carry_kernel<<<(B * D) / 256, 256, 0, stream>>>(S, a_log, carry, D);
  scan_apply_kernel<<<dim3(D / 256, CHUNKS, B), 256, 0, stream>>>(
      x, normed, buf1, carry, a_log, c_out, d_skip, x1, T, D);

  rmsnorm_kernel<<<M, 256, 0, stream>>>(x1, ffn_w, buf1 /*n2*/, D);

  // hidden = silu(n2 @ fc^T)   (M x H, K = D)
  gemm_wmma_f32<<<dim3(H / 128, M / 128), 256, 0, stream>>>(
      buf1, fc_w, nullptr, hidden, M, D, H, 1);
  // out = x1 + hidden @ proj^T (M x D, K = H)
  gemm_wmma_f32<<<dim3(D / 128, M / 128), 256, 0, stream>>>(
      hidden, proj_w, x1, out, M, H, D, 0);
}